// _BiDirectionalBlock_42580305772940
// MI455X (gfx1250) — compile-verified
//
#include <hip/hip_runtime.h>
#include <math.h>

#define N_NODES 50000
#define E_EDGES 800000
#define FIN     256
#define FOUT    256
#define HEADS   4
#define HALF    128
#define DH      32
#define PCOLS   512        // fused projection columns: q_in|kv_in|q_out|kv_out
#define TAUF    0.25f
#define EPSF    1e-9f
#define BN_EPSF 1e-5f

typedef __attribute__((ext_vector_type(2))) float v2f;
typedef __attribute__((ext_vector_type(8))) float v8f;

// ---------------------------------------------------------------------------
// Weight packing: [HEADS, FIN, DH] x4 -> row-major [K=256, 512]
// col c: dir=c>>8, which=(c>>7)&1, j=c&127, h=j>>5, d=j&31
// ---------------------------------------------------------------------------
__global__ void pack_proj_kernel(const float* __restrict__ Wq_in,
                                 const float* __restrict__ Wkv_in,
                                 const float* __restrict__ Wq_out,
                                 const float* __restrict__ Wkv_out,
                                 float* __restrict__ Wpack) {
    int idx = blockIdx.x * blockDim.x + threadIdx.x;   // FIN * PCOLS
    if (idx >= FIN * PCOLS) return;
    int f = idx >> 9;
    int c = idx & (PCOLS - 1);
    int dir = c >> 8, which = (c >> 7) & 1, j = c & 127;
    int h = j >> 5, d = j & 31;
    const float* W = (dir == 0) ? ((which == 0) ? Wq_in : Wkv_in)
                                : ((which == 0) ? Wq_out : Wkv_out);
    Wpack[f * PCOLS + c] = W[(h * FIN + f) * DH + d];
}

__global__ void pack_lin_kernel(const float* __restrict__ linW,
                                float* __restrict__ WpackL) {
    int idx = blockIdx.x * blockDim.x + threadIdx.x;   // FOUT * FOUT
    if (idx >= FOUT * FOUT) return;
    int f = idx >> 8;
    int o = idx & 255;
    WpackL[f * FOUT + o] = linW[o * FOUT + f];         // transpose
}

// ---------------------------------------------------------------------------
// fp32 WMMA GEMM: C[N,OC] = A[N,256] * B[256,OC] (+ bias[OC])
// OC is a compile-time constant -> all B loads / C stores use immediate
// ioffsets off a single base pointer (no per-iteration 64-bit address math).
// block = 128 threads (4 waves); wave w computes 16x16 tile at col by*64+w*16
// A tile (16 rows x 256) staged in LDS with 264-float padded rows.
// V_WMMA_F32_16X16X4_F32 fragment layout (ISA 7.12.2):
//   A: lane holds row M=lane%16, K = 2*(lane/16)+{0,1}
//   B: lane holds col N=lane%16, K = 2*(lane/16)+{0,1}
//   C/D: lane%16 = N, VGPR v -> M = v + 8*(lane/16)
// ---------------------------------------------------------------------------
template <int OC, bool HAS_BIAS>
__global__ __launch_bounds__(128) void wmma_gemm_kernel(
        const float* __restrict__ A, const float* __restrict__ B,
        const float* __restrict__ bias, float* __restrict__ C) {
    __shared__ float lds[16 * 264];
    const int t    = threadIdx.x;
    const int wave = t >> 5;
    const int lane = t & 31;
    const int m    = lane & 15;
    const int g    = lane >> 4;
    const long rowBase = (long)blockIdx.x * 16;
    const float* Arow = A + rowBase * FIN;

    // Stage 16x256 A tile (coalesced float4 loads)
    #pragma unroll
    for (int i = 0; i < 8; ++i) {
        int q   = t + i * 128;          // float4 index in [0,1024)
        int row = q >> 6;               // 64 float4 per row
        int c4  = (q & 63) << 2;
        float4 v = *(const float4*)(Arow + row * FIN + c4);
        *(float4*)&lds[row * 264 + c4] = v;
    }
    __syncthreads();

    const int col = blockIdx.y * 64 + wave * 16 + m;
    const float* __restrict__ Bcol = B + col;          // single base; k offsets immediate
    const float* __restrict__ Alds = &lds[m * 264 + 2 * g];
    v8f acc = {};
    #pragma unroll
    for (int kb = 0; kb < 64; ++kb) {
        const int k0 = kb * 4 + 2 * g;                 // lane's K pair
        v2f a, b;
        a.x = Alds[kb * 4];
        a.y = Alds[kb * 4 + 1];
        b.x = Bcol[(kb * 4 + 2 * g) * OC];
        b.y = Bcol[(kb * 4 + 2 * g + 1) * OC];
        (void)k0;
        acc = __builtin_amdgcn_wmma_f32_16x16x4_f32(
                  false, a, false, b, (short)0, acc, false, false);
    }
    float bv = HAS_BIAS ? bias[col] : 0.0f;
    float* __restrict__ Cbase = C + (rowBase + 8 * g) * OC + col;
    #pragma unroll
    for (int v = 0; v < 8; ++v)
        Cbase[v * OC] = acc[v] + bv;
}

// ---------------------------------------------------------------------------
// Per-(node, head-group) L2 norms over 32 contiguous floats of P row.
// group g in [0,16): 0-3 q_in heads, 4-7 kv_in, 8-11 q_out, 12-15 kv_out
// ---------------------------------------------------------------------------
__global__ void norms_kernel(const float* __restrict__ P,
                             float* __restrict__ norms) {
    int tid = blockIdx.x * blockDim.x + threadIdx.x;
    if (tid >= N_NODES * 16) return;
    int n = tid >> 4, g = tid & 15;
    const float* p = P + (long)n * PCOLS + g * 32;
    float s = 0.f;
    #pragma unroll
    for (int i = 0; i < 32; i += 4) {
        float4 v = *(const float4*)(p + i);
        s += v.x * v.x + v.y * v.y + v.z * v.z + v.w * v.w;
    }
    norms[tid] = sqrtf(s) + EPSF;
}

// Hcat init with per-direction biases (scatter target must start at bias)
__global__ void init_hcat_kernel(const float* __restrict__ bias_in,
                                 const float* __restrict__ bias_out,
                                 float* __restrict__ Hcat) {
    long tid = (long)blockIdx.x * blockDim.x + threadIdx.x;
    if (tid >= (long)N_NODES * FOUT) return;
    int j = (int)(tid & 255);
    Hcat[tid] = (j < HALF) ? bias_in[j] : bias_out[j - HALF];
}

// ---------------------------------------------------------------------------
// Edge pass 1 — one wave32 per edge.
// Lane L loads float4 #L of the 128-float q/kv rows (fully coalesced; two
// cachelines per row).  Head h owns lanes [8h,8h+8); 3 shfl_xor steps give the
// per-head 32-element dot.  Group leaders (lane%8==0) compute the score and do
// the coalesced denom atomic.
// ---------------------------------------------------------------------------
__global__ __launch_bounds__(256) void edge_pass1_kernel(
        const int* __restrict__ edge, const float* __restrict__ P,
        const float* __restrict__ norms, const float* __restrict__ lvec,
        const float* __restrict__ cptr, const float* __restrict__ dptr,
        int qofs, int kvofs,
        float* __restrict__ exps, float* __restrict__ denom) {
    int wid  = (blockIdx.x * blockDim.x + threadIdx.x) >> 5;  // edge id
    int lane = threadIdx.x & 31;
    if (wid >= E_EDGES) return;
    int e   = wid;
    int src = edge[e];
    int dst = edge[E_EDGES + e];
    const float4* q  = (const float4*)(P + (long)dst * PCOLS + qofs);
    const float4* kv = (const float4*)(P + (long)src * PCOLS + kvofs);
    float4 a = q[lane];
    float4 b = kv[lane];
    float part = a.x * b.x + a.y * b.y + a.z * b.z + a.w * b.w;
    part += __shfl_xor(part, 1);
    part += __shfl_xor(part, 2);
    part += __shfl_xor(part, 4);           // lanes in 8-group hold head dot
    int h = lane >> 3;
    if ((lane & 7) == 0) {
        float nq   = norms[(long)dst * 16 + (qofs  >> 5) + h];
        float nkv  = norms[(long)src * 16 + (kvofs >> 5) + h];
        float cosv = fabsf(part / (nq * nkv));
        float z    = cptr[0] * lvec[src] + dptr[0];
        float Dj   = 1.0f / (1.0f + expf(-z));
        float ex   = expf(cosv * Dj / TAUF);
        exps[(long)e * HEADS + h] = ex;
        unsafeAtomicAdd(&denom[(long)dst * HEADS + h], ex);
    }
}

// ---------------------------------------------------------------------------
// Edge pass 2 — one wave32 per edge.  Output element j = i*32 + lane (head i):
// kv loads and the scatter atomics are perfectly lane-coalesced per sweep.
// ---------------------------------------------------------------------------
__global__ __launch_bounds__(256) void edge_pass2_kernel(
        const int* __restrict__ edge, const float* __restrict__ P,
        const float* __restrict__ exps, const float* __restrict__ denom,
        float* __restrict__ Hcat, int kvofs, int outofs) {
    int wid  = (blockIdx.x * blockDim.x + threadIdx.x) >> 5;  // edge id
    int lane = threadIdx.x & 31;
    if (wid >= E_EDGES) return;
    int e   = wid;
    int src = edge[e];
    int dst = edge[E_EDGES + e];
    const float* kv = P + (long)src * PCOLS + kvofs;
    float* o = Hcat + (long)dst * FOUT + outofs;
    #pragma unroll
    for (int i = 0; i < HEADS; ++i) {
        float alpha = exps[(long)e * HEADS + i] /
                      (denom[(long)dst * HEADS + i] + EPSF);
        unsafeAtomicAdd(&o[i * 32 + lane], alpha * kv[i * 32 + lane]);
    }
}

// ---------------------------------------------------------------------------
// BatchNorm batch statistics + exact-erf GELU
// ---------------------------------------------------------------------------
__global__ __launch_bounds__(256) void bn_partial_kernel(
        const float* __restrict__ H, float* __restrict__ bnsum,
        float* __restrict__ bnsq) {
    int t = threadIdx.x;                     // feature id
    long r0 = (long)blockIdx.x * 64;
    float s = 0.f, q = 0.f;
    for (int i = 0; i < 64; ++i) {
        long r = r0 + i;
        if (r < N_NODES) {
            float v = H[r * FOUT + t];
            s += v; q += v * v;
        }
    }
    unsafeAtomicAdd(&bnsum[t], s);
    unsafeAtomicAdd(&bnsq[t], q);
}

__global__ void bn_finalize_kernel(const float* __restrict__ bnsum,
                                   const float* __restrict__ bnsq,
                                   const float* __restrict__ gamma,
                                   const float* __restrict__ beta,
                                   float* __restrict__ scale,
                                   float* __restrict__ shift) {
    int t = threadIdx.x;
    float mean = bnsum[t] / (float)N_NODES;
    float var  = bnsq[t] / (float)N_NODES - mean * mean;  // biased var
    float inv  = rsqrtf(var + BN_EPSF);
    float sc   = gamma[t] * inv;
    scale[t] = sc;
    shift[t] = beta[t] - mean * sc;
}

__global__ void bn_gelu_kernel(const float* __restrict__ H,
                               const float* __restrict__ scale,
                               const float* __restrict__ shift,
                               float* __restrict__ out) {
    long tid = (long)blockIdx.x * blockDim.x + threadIdx.x;
    if (tid >= (long)N_NODES * FOUT) return;
    int c = (int)(tid & 255);
    float y = H[tid] * scale[c] + shift[c];
    out[tid] = 0.5f * y * (1.0f + erff(y * 0.70710678118654752440f));
}

// ---------------------------------------------------------------------------
extern "C" void kernel_launch(void* const* d_in, const int* in_sizes, int n_in,
                              void* d_out, int out_size, void* d_ws, size_t ws_size,
                              hipStream_t stream) {
    const float* x        = (const float*)d_in[0];
    const int*   edge_in  = (const int*)  d_in[1];
    const int*   edge_out = (const int*)  d_in[2];
    const float* l_vec    = (const float*)d_in[3];
    const float* Wq_in    = (const float*)d_in[4];
    const float* Wkv_in   = (const float*)d_in[5];
    const float* bias_in  = (const float*)d_in[6];
    const float* c_in     = (const float*)d_in[7];
    const float* dd_in    = (const float*)d_in[8];
    const float* Wq_out   = (const float*)d_in[9];
    const float* Wkv_out  = (const float*)d_in[10];
    const float* bias_out = (const float*)d_in[11];
    const float* c_out    = (const float*)d_in[12];
    const float* dd_out   = (const float*)d_in[13];
    const float* lin_W    = (const float*)d_in[14];
    const float* lin_b    = (const float*)d_in[15];
    const float* bn_gamma = (const float*)d_in[16];
    const float* bn_beta  = (const float*)d_in[17];
    float* out = (float*)d_out;

    // workspace layout (bytes)
    char* ws = (char*)d_ws;
    size_t off = 0;
    float* P       = (float*)(ws + off); off += (size_t)N_NODES * PCOLS * 4;  // 102.4 MB
    float* norms   = (float*)(ws + off); off += (size_t)N_NODES * 16 * 4;
    float* denomIn = (float*)(ws + off); off += (size_t)N_NODES * HEADS * 4;
    float* denomOut= (float*)(ws + off); off += (size_t)N_NODES * HEADS * 4;
    float* expsIn  = (float*)(ws + off); off += (size_t)E_EDGES * HEADS * 4;
    float* expsOut = (float*)(ws + off); off += (size_t)E_EDGES * HEADS * 4;
    float* Hcat    = (float*)(ws + off); off += (size_t)N_NODES * FOUT * 4;
    float* Wpack   = (float*)(ws + off); off += (size_t)FIN * PCOLS * 4;
    float* WpackL  = (float*)(ws + off); off += (size_t)FOUT * FOUT * 4;
    float* bnsum   = (float*)(ws + off); off += 256 * 4;
    float* bnsq    = (float*)(ws + off); off += 256 * 4;
    float* bnscale = (float*)(ws + off); off += 256 * 4;
    float* bnshift = (float*)(ws + off); off += 256 * 4;
    float* Hlin    = P;  // P is dead after edge pass 2; reuse for linear output

    // zero accumulators (denomIn+denomOut contiguous; bnsum+bnsq contiguous)
    hipMemsetAsync(denomIn, 0, (size_t)N_NODES * HEADS * 4 * 2, stream);
    hipMemsetAsync(bnsum, 0, 256 * 4 * 2, stream);

    // 1) pack weights
    pack_proj_kernel<<<(FIN * PCOLS + 255) / 256, 256, 0, stream>>>(
        Wq_in, Wkv_in, Wq_out, Wkv_out, Wpack);
    pack_lin_kernel<<<(FOUT * FOUT + 255) / 256, 256, 0, stream>>>(lin_W, WpackL);

    // 2) fused projections: P[N,512] = x @ Wpack  (fp32 WMMA)
    {
        dim3 grid(N_NODES / 16, PCOLS / 64);
        wmma_gemm_kernel<PCOLS, false><<<grid, 128, 0, stream>>>(x, Wpack, nullptr, P);
    }

    // 3) per-(node,head) norms
    norms_kernel<<<(N_NODES * 16 + 255) / 256, 256, 0, stream>>>(P, norms);

    // 4) Hcat = concat biases (scatter target)
    init_hcat_kernel<<<(N_NODES * FOUT + 255) / 256, 256, 0, stream>>>(
        bias_in, bias_out, Hcat);

    // 5) edge softmax numerators + denominators (wave per edge, both dirs)
    const int egrid = (E_EDGES * 32 + 255) / 256;   // one wave32 per edge
    edge_pass1_kernel<<<egrid, 256, 0, stream>>>(
        edge_in, P, norms, l_vec, c_in, dd_in, /*qofs=*/0, /*kvofs=*/128,
        expsIn, denomIn);
    edge_pass1_kernel<<<egrid, 256, 0, stream>>>(
        edge_out, P, norms, l_vec, c_out, dd_out, /*qofs=*/256, /*kvofs=*/384,
        expsOut, denomOut);

    // 6) alpha-weighted message aggregation (wave per edge)
    edge_pass2_kernel<<<egrid, 256, 0, stream>>>(
        edge_in, P, expsIn, denomIn, Hcat, /*kvofs=*/128, /*outofs=*/0);
    edge_pass2_kernel<<<egrid, 256, 0, stream>>>(
        edge_out, P, expsOut, denomOut, Hcat, /*kvofs=*/384, /*outofs=*/128);

    // 7) linear: Hlin = Hcat @ lin_W^T + lin_b  (fp32 WMMA)
    {
        dim3 grid(N_NODES / 16, FOUT / 64);
        wmma_gemm_kernel<FOUT, true><<<grid, 128, 0, stream>>>(Hcat, WpackL, lin_b, Hlin);
    }

    // 8) batch-norm stats + affine + exact GELU
    bn_partial_kernel<<<(N_NODES + 63) / 64, 256, 0, stream>>>(Hlin, bnsum, bnsq);
    bn_finalize_kernel<<<1, 256, 0, stream>>>(bnsum, bnsq, bn_gamma, bn_beta,
                                              bnscale, bnshift);
    bn_gelu_kernel<<<(N_NODES * FOUT + 255) / 256, 256, 0, stream>>>(
        Hlin, bnscale, bnshift, out);
}